// ElecWholePoseScoringModule_20194936225935
// MI455X (gfx1250) — compile-verified
//
#include <hip/hip_runtime.h>
#include <hip/hip_bf16.h>

// v2f = f32 16x4 A/B fragment (2 VGPRs/lane), v8f = f32 16x16 C/D fragment.
typedef float v2f __attribute__((ext_vector_type(2)));
typedef float v8f __attribute__((ext_vector_type(8)));
typedef float v4f __attribute__((ext_vector_type(4)));

#define A_MAX   2048   // max padded atoms per pose staged in LDS
#define C_MAX   4      // max inter-block connections supported
#define NSPLIT  16     // workgroups per pose (spread tile-pairs over WGPs)
#define BLOCK   256    // 8 wave32s per workgroup

__global__ __launch_bounds__(BLOCK)
void elec_pose_wmma_kernel(const float* __restrict__ coords,
                           const int*   __restrict__ offsets,        // (P,B)
                           const int*   __restrict__ block_types,    // (P,B)
                           const int*   __restrict__ inter_bondsep,  // (P,B,B,C,C)
                           const int*   __restrict__ bt_n_atoms,     // (NBT)
                           const float* __restrict__ bt_partial_charge, // (NBT,APB)
                           const int*   __restrict__ bt_inter_pd,    // (NBT,C,APB)
                           const int*   __restrict__ bt_intra_pd,    // (NBT,APB,APB)
                           const float* __restrict__ gparams,        // (1,5)
                           float*       __restrict__ partials,       // (P,NSPLIT)
                           int P, int A, int B, int APB, int C)
{
    __shared__ float sX[A_MAX], sY[A_MAX], sZ[A_MAX], sSq[A_MAX], sQ[A_MAX];
    __shared__ int   sMeta[A_MAX];  // blk | aw<<8 | bt<<16 | valid<<24
    __shared__ int   sD2C[A_MAX];   // per-connection path dist, 8 bits each
    __shared__ float sWave[BLOCK / 32];

    const int p     = blockIdx.x / NSPLIT;
    const int split = blockIdx.x % NSPLIT;
    const int tid   = threadIdx.x;

    if (A > A_MAX) A = A_MAX;                 // safety clamp (compile-time cap)
    const int Cc = (C < C_MAX) ? C : C_MAX;

    const float D = gparams[0], D0 = gparams[1], S = gparams[2];
    const float min_dis = gparams[3], max_dis = gparams[4];
    const float mds = max_dis * S;
    const float eps_max = D - 0.5f * (D - D0) * (2.f + 2.f * mds + mds * mds) * __expf(-mds);
    const float shift = 1.f / (max_dis * eps_max);

    const int nTiles = (A + 15) >> 4;
    const int Apad   = nTiles << 4;

    // ---- Stage per-atom data into LDS (general searchsorted block map) ----
    const int* offs = offsets + (size_t)p * B;
    for (int i = tid; i < Apad; i += blockDim.x) {
        if (i < A) {
            int lo = 0, hi = B;                       // searchsorted right
            while (lo < hi) { int mid = (lo + hi) >> 1;
                              if (offs[mid] <= i) lo = mid + 1; else hi = mid; }
            int blk = lo - 1; blk = (blk < 0) ? 0 : ((blk > B - 1) ? B - 1 : blk);
            int aw  = i - offs[blk];
            int bt  = block_types[(size_t)p * B + blk];
            int valid = (aw < bt_n_atoms[bt]) ? 1 : 0;
            int awc = aw; awc = (awc < 0) ? 0 : ((awc > APB - 1) ? APB - 1 : awc);
            float x = coords[((size_t)p * A + i) * 3 + 0];
            float y = coords[((size_t)p * A + i) * 3 + 1];
            float z = coords[((size_t)p * A + i) * 3 + 2];
            sX[i] = x; sY[i] = y; sZ[i] = z;
            sSq[i] = x * x + y * y + z * z;
            sQ[i]  = bt_partial_charge[(size_t)bt * APB + awc];
            sMeta[i] = (blk & 255) | ((awc & 255) << 8) | ((bt & 255) << 16) | (valid << 24);
            int pk = 0;
            for (int c = 0; c < Cc; ++c) {
                int d = bt_inter_pd[((size_t)bt * C + c) * APB + awc];
                d = (d < 0) ? 0 : ((d > 63) ? 63 : d);
                pk |= d << (8 * c);
            }
            sD2C[i] = pk;
        } else {
            sX[i] = 0.f; sY[i] = 0.f; sZ[i] = 0.f; sSq[i] = 0.f; sQ[i] = 0.f;
            sMeta[i] = 0; sD2C[i] = 0;
        }
    }
    __syncthreads();

    const int lane   = tid & 31;
    const int wid    = tid >> 5;
    const int nWaves = blockDim.x >> 5;
    const int hi16   = lane >> 4;        // 0: K=0,1 half; 1: K=2,3 half
    const int ln     = lane & 15;

    // Loop-invariant fragment source selection: avoids per-tile exec-mask
    // branching (single divergent-address ds_load + cndmask instead).
    const float* __restrict__ sK0 = hi16 ? sZ : sX;   // K=0 (lo) / K=2 (hi)
    const float  yMask = hi16 ? 0.f : 1.f;            // K=1 (lo) / K=3=pad (hi)

    const int totalPairs = nTiles * (nTiles + 1) / 2;
    const int gWave   = split * nWaves + wid;
    const int gStride = NSPLIT * nWaves;

    float acc = 0.f;

    for (int k = gWave; k < totalPairs; k += gStride) {
        // decode upper-triangle tile pair (ti <= tj) — uniform across the wave
        int kk = k, ti = 0;
        while (kk >= nTiles - ti) { kk -= (nTiles - ti); ++ti; }
        const int tj = ti + kk;
        const int i0 = ti << 4, j0 = tj << 4;

        // A fragment (rows i0..i0+15 x K=x,y,z,0), B fragment (K x cols j0..)
        const int ia = i0 + ln, ja = j0 + ln;
        v2f afrag, bfrag;
        afrag[0] = sK0[ia];
        afrag[1] = yMask * sY[ia];
        bfrag[0] = sK0[ja];
        bfrag[1] = yMask * sY[ja];

        v8f cm = {0.f, 0.f, 0.f, 0.f, 0.f, 0.f, 0.f, 0.f};
        // 16x16 tile of dot products xi . xj in one matrix op
        cm = __builtin_amdgcn_wmma_f32_16x16x4_f32(false, afrag, false, bfrag,
                                                   (short)0, cm, false, false);

        // ---- j-side data: fixed across the 8 C-elements of this lane ----
        const int j   = j0 + ln;
        const int mj  = sMeta[j];
        if (!(mj & (1 << 24))) continue;            // invalid col atom: all zero
        const float sqj = sSq[j];
        const float qj  = sQ[j];
        const int   dj  = sD2C[j];
        const int   blkj = mj & 255;
        const int   awj  = (mj >> 8) & 255;

        // ---- i-side |x|^2: 8 consecutive values -> two ds_load_b128 ----
        const int ib = i0 + (hi16 ? 8 : 0);
        float sqi[8];
        *(v4f*)&sqi[0] = *(const v4f*)&sSq[ib];
        *(v4f*)&sqi[4] = *(const v4f*)&sSq[ib + 4];

        #pragma unroll
        for (int v = 0; v < 8; ++v) {
            const int i = ib + v;
            if (i >= j) continue;                    // diagonal tile: strict triu
            // cheap distance reject first (~5% survive for 25A box, cutoff 5.5A)
            const float d2   = sqi[v] + sqj - 2.f * cm[v];
            const float dist = sqrtf(fmaxf(d2, 1e-12f));
            if (dist >= max_dis) continue;

            const int mi = sMeta[i];                 // lazy i-side metadata
            if (!(mi & (1 << 24))) continue;

            const int blki = mi & 255;
            int bpl;
            if (blki == blkj) {
                const int bti = (mi >> 16) & 255;
                const int awi = (mi >> 8) & 255;
                bpl = bt_intra_pd[((size_t)bti * APB + awi) * APB + awj];
            } else {
                const int* bs = inter_bondsep +
                    ((((size_t)p * B + blki) * B + blkj) * C) * C;
                const int di = sD2C[i];
                bpl = 0x7fffffff;
                for (int c1 = 0; c1 < Cc; ++c1)
                    for (int c2 = 0; c2 < Cc; ++c2) {
                        int s = ((di >> (8 * c1)) & 255) + bs[c1 * C + c2]
                              + ((dj >> (8 * c2)) & 255);
                        bpl = (s < bpl) ? s : bpl;
                    }
            }
            const float cp = (bpl > 4) ? 1.f : ((bpl == 4) ? 0.2f : 0.f);
            if (cp == 0.f) continue;

            const float dc  = fminf(fmaxf(dist, min_dis), max_dis);
            const float ds  = dc * S;
            const float eps = D - 0.5f * (D - D0) * (2.f + 2.f * ds + ds * ds) * __expf(-ds);
            const float e   = 322.0637f * sQ[i] * qj * (1.f / (dc * eps) - shift);
            acc += cp * e;
        }
    }

    // ---- deterministic reduction: shuffle tree within wave, fixed-order sum across waves ----
    for (int off = 16; off > 0; off >>= 1)
        acc += __shfl_xor(acc, off, 32);
    if (lane == 0) sWave[wid] = acc;
    __syncthreads();
    if (tid == 0) {
        float s = 0.f;
        for (int w = 0; w < nWaves; ++w) s += sWave[w];
        partials[(size_t)p * NSPLIT + split] = s;
    }
}

__global__ void elec_pose_reduce_kernel(const float* __restrict__ partials,
                                        float* __restrict__ out, int P)
{
    const int p = blockIdx.x * blockDim.x + threadIdx.x;
    if (p < P) {
        float s = 0.f;
        for (int i = 0; i < NSPLIT; ++i) s += partials[(size_t)p * NSPLIT + i];
        out[p] = s;
    }
}

extern "C" void kernel_launch(void* const* d_in, const int* in_sizes, int n_in,
                              void* d_out, int out_size, void* d_ws, size_t ws_size,
                              hipStream_t stream)
{
    (void)n_in; (void)ws_size;
    // Derive shapes from flat element counts (setup_inputs dict order).
    const int NBT = in_sizes[5];
    const int APB = in_sizes[6] / NBT;
    const int PB  = in_sizes[1];                 // P*B
    const int B   = in_sizes[3] / PB;            // (P*B*B)/(P*B)
    const int P   = PB / B;
    const int A   = in_sizes[0] / (P * 3);
    const int C   = in_sizes[9] / (NBT * APB);

    const float* coords       = (const float*)d_in[0];
    const int*   offsets      = (const int*)  d_in[1];
    const int*   block_types  = (const int*)  d_in[2];
    const int*   inter_bsep   = (const int*)  d_in[4];
    const int*   bt_n_atoms   = (const int*)  d_in[5];
    const float* bt_q         = (const float*)d_in[6];
    const int*   bt_inter_pd  = (const int*)  d_in[9];
    const int*   bt_intra_pd  = (const int*)  d_in[10];
    const float* gparams      = (const float*)d_in[11];
    float*       out          = (float*)d_out;
    float*       partials     = (float*)d_ws;    // P*NSPLIT floats

    elec_pose_wmma_kernel<<<P * NSPLIT, BLOCK, 0, stream>>>(
        coords, offsets, block_types, inter_bsep, bt_n_atoms, bt_q,
        bt_inter_pd, bt_intra_pd, gparams, partials, P, A, B, APB, C);

    elec_pose_reduce_kernel<<<(P + 63) / 64, 64, 0, stream>>>(partials, out, P);
    (void)out_size;
}